// DipolePredictorE3NN_72043781423353
// MI455X (gfx1250) — compile-verified
//
#include <hip/hip_runtime.h>

typedef float v2f __attribute__((ext_vector_type(2)));
typedef float v8f __attribute__((ext_vector_type(8)));

// Pointer types for the async global->LDS builtin:
//   param0: int4 addrspace(1)* (prints as "__device__" in HIP diags)
//   param1: int4 addrspace(3)* (LDS)
typedef int v4i __attribute__((vector_size(16)));
typedef __attribute__((address_space(1))) v4i g_v4i;
typedef __attribute__((address_space(3))) v4i l_v4i;

#define B_DIM 1024
#define N_DIM 512
#define NS    7
#define HID   128

// alpha = 1/sqrt(7); ge = alpha * mean_n => scale = alpha / N
#define GE_SCALE (0.3779644730092272f / 512.0f)

#define FEAT_F4  (N_DIM * NS / 4)        // 896 float4 of feats per batch
#define COOR_F4  (N_DIM * 3 / 4)         // 384 float4 of coors per batch
#define TILE_F4  (FEAT_F4 + COOR_F4)     // 1280 float4 = 20480 B per batch

// ---------------------------------------------------------------------------
// Kernel 1: ge[b] = (alpha/N) * sum_n <feats[b,n],W_tp> * coors[b,n]
// One 256-thread block per batch. The batch tile (feats 14336B + coors 6144B,
// both 16B-aligned) is staged into LDS with async b128 copies (ASYNCcnt path),
// then each thread computes 2 points from LDS (strides 7 & 3 are coprime with
// 64 banks -> conflict-free), wave32 butterfly-reduce, ds_add_f32, write ge.
// adj_mat (268 MB) is never touched.
// ---------------------------------------------------------------------------
__global__ __launch_bounds__(256) void dipole_reduce_kernel(
    const float* __restrict__ feats,   // [B, N, 7]
    const float* __restrict__ coors,   // [B, N, 3]
    const float* __restrict__ W_tp,    // [7]
    float* __restrict__ ge)            // [B, 3]  (workspace)
{
    __shared__ float4 stage4[TILE_F4];             // 20480 B
    __shared__ float  acc[3];
    float* stage = (float*)stage4;                 // feats [0,3584), coors [3584,5120)

    const int b = blockIdx.x;
    const int t = threadIdx.x;
    if (t < 3) acc[t] = 0.0f;

    const float4* fsrc = (const float4*)(feats + (size_t)b * (N_DIM * NS));
    const float4* csrc = (const float4*)(coors + (size_t)b * (N_DIM * 3));

    // ---- stage tile into LDS: 1280 async b128 transfers, 5 per thread ----
    #pragma unroll
    for (int k = 0; k < TILE_F4 / 256; ++k) {
        const int i = t + k * 256;
        const float4* src = (i < FEAT_F4) ? (fsrc + i) : (csrc + (i - FEAT_F4));
#if __has_builtin(__builtin_amdgcn_global_load_async_to_lds_b128)
        __builtin_amdgcn_global_load_async_to_lds_b128(
            (g_v4i*)src, (l_v4i*)&stage4[i], /*offset=*/0, /*cpol=*/0);
#else
        stage4[i] = *src;                          // fallback: b128 load + ds_store_b128
#endif
    }

#if __has_builtin(__builtin_amdgcn_global_load_async_to_lds_b128)
  #if __has_builtin(__builtin_amdgcn_s_wait_asynccnt)
    __builtin_amdgcn_s_wait_asynccnt(0);
  #else
    asm volatile("s_wait_asynccnt 0x0" ::: "memory");
  #endif
#endif
    __syncthreads();

    // Uniform weights -> scalar loads (issued alongside the copies)
    const float w0 = W_tp[0], w1 = W_tp[1], w2 = W_tp[2], w3 = W_tp[3];
    const float w4 = W_tp[4], w5 = W_tp[5], w6 = W_tp[6];

    const float* fl = stage;                       // [512][7]
    const float* cl = stage + N_DIM * NS;          // [512][3]

    float ax = 0.0f, ay = 0.0f, az = 0.0f;
    #pragma unroll
    for (int r = 0; r < 2; ++r) {
        const int n = t + r * 256;
        const float* fr = fl + n * NS;
        const float* cr = cl + n * 3;
        const float s = fr[0]*w0 + fr[1]*w1 + fr[2]*w2 + fr[3]*w3
                      + fr[4]*w4 + fr[5]*w5 + fr[6]*w6;
        ax += s * cr[0];
        ay += s * cr[1];
        az += s * cr[2];
    }

    // wave32 butterfly reduction
    #pragma unroll
    for (int m = 16; m >= 1; m >>= 1) {
        ax += __shfl_xor(ax, m, 32);
        ay += __shfl_xor(ay, m, 32);
        az += __shfl_xor(az, m, 32);
    }
    if ((t & 31) == 0) {                           // one lane per wave -> ds_add_f32
        atomicAdd(&acc[0], ax);
        atomicAdd(&acc[1], ay);
        atomicAdd(&acc[2], az);
    }
    __syncthreads();

    if (t < 3) ge[(size_t)b * 3 + t] = acc[t] * GE_SCALE;
}

// ---------------------------------------------------------------------------
// Kernel 2: out = relu(ge @ W1^T + b1) @ W2^T + b2   via V_WMMA_F32_16X16X4_F32
// 4 waves/block, one 16-row batch tile per wave. All operand loads are
// branchless (clamped indices + multiplicative masks) so EXEC stays all-ones
// with no save/restore around the WMMAs. h staged in LDS with 132-float pitch
// (pitch%64banks = 4 -> conflict-free layer-2 A reloads).
// ---------------------------------------------------------------------------
__global__ __launch_bounds__(128) void dipole_mlp_wmma_kernel(
    const float* __restrict__ ge,   // [B, 3]
    const float* __restrict__ W1,   // [128, 3] row-major
    const float* __restrict__ b1,   // [128]
    const float* __restrict__ W2,   // [3, 128] row-major
    const float* __restrict__ b2,   // [3]
    float* __restrict__ out)        // [B, 3]
{
    constexpr int HSTR = HID + 4;                  // 132
    __shared__ float hbuf[4][16][HSTR];            // 33,792 B

    const int wave = threadIdx.x >> 5;
    const int lane = threadIdx.x & 31;
    const int half = lane >> 4;                    // 0: lanes 0-15, 1: lanes 16-31
    const int l16  = lane & 15;
    const int b0   = (blockIdx.x * 4 + wave) * 16;
    const float hm = half ? 0.0f : 1.0f;           // zeroes the padded K=3 slot

    // ---- A tile (16x4, K=3 valid): lanes 0-15 hold K=0/1, lanes 16-31 K=2/3
    v2f a1;
    {
        const int g  = (b0 + l16) * 3 + half * 2;  // always < 3072
        int gy = g + 1;
        gy = gy > (B_DIM * 3 - 1) ? (B_DIM * 3 - 1) : gy;   // clamp (value masked)
        a1.x = ge[g];
        a1.y = ge[gy] * hm;
    }

    // ---- Layer 1: 8 N-tiles of 16 over HID=128
    #pragma unroll
    for (int j = 0; j < 8; ++j) {
        const int n  = j * 16 + l16;               // hidden-unit column
        const int wb = n * 3 + half * 2;           // B[k][n] = W1[n][k]; always < 384
        int wy = wb + 1;
        wy = wy > (HID * 3 - 1) ? (HID * 3 - 1) : wy;
        v2f bm;
        bm.x = W1[wb];
        bm.y = W1[wy] * hm;

        v8f cz = {};
        v8f d = __builtin_amdgcn_wmma_f32_16x16x4_f32(
                    /*neg_a=*/false, a1, /*neg_b=*/false, bm,
                    /*c_mod=*/(short)0, cz, /*reuse_a=*/false, /*reuse_b=*/false);

        const float bias = b1[n];
        #pragma unroll
        for (int r = 0; r < 8; ++r) {              // D: VGPR r -> row r + 8*half
            const int M = r + half * 8;
            float h = d[r] + bias;
            h = h > 0.0f ? h : 0.0f;               // ReLU
            hbuf[wave][M][n] = h;
        }
    }
    __syncthreads();

    // ---- Layer 2: out = h @ W2^T, K=128 in 32 steps of 4, accumulate in C
    const float nmask = (l16 < 3) ? 1.0f : 0.0f;   // pad N=3 -> 16, branchless
    const int   ncl   = (l16 < 3) ? l16 : 0;       // clamped W2 row

    v8f c2 = {};
    #pragma unroll 4
    for (int kk = 0; kk < 32; ++kk) {
        const int kb = kk * 4 + half * 2;
        v2f a2, bm2;
        a2.x = hbuf[wave][l16][kb + 0];
        a2.y = hbuf[wave][l16][kb + 1];
        bm2.x = W2[ncl * HID + kb + 0] * nmask;    // B[k][n] = W2[n][k] (masked)
        bm2.y = W2[ncl * HID + kb + 1] * nmask;
        c2 = __builtin_amdgcn_wmma_f32_16x16x4_f32(
                 false, a2, false, bm2, (short)0, c2, false, false);
    }

    // ---- Epilogue: bias + store (divergence only after all WMMAs)
    if (l16 < 3) {
        const float bout = b2[l16];
        #pragma unroll
        for (int r = 0; r < 8; ++r) {
            const int M = r + half * 8;
            out[(size_t)(b0 + M) * 3 + l16] = c2[r] + bout;
        }
    }
}

// ---------------------------------------------------------------------------
// Launch: inputs (feats, coors, adj_mat, W_tp, W1, b1, W2, b2); adj_mat unused.
// Output [1024,3] f32. Workspace holds ge [1024,3], fully overwritten per call.
// ---------------------------------------------------------------------------
extern "C" void kernel_launch(void* const* d_in, const int* in_sizes, int n_in,
                              void* d_out, int out_size, void* d_ws, size_t ws_size,
                              hipStream_t stream) {
    (void)in_sizes; (void)n_in; (void)out_size; (void)ws_size;

    const float* feats = (const float*)d_in[0];
    const float* coors = (const float*)d_in[1];
    /* d_in[2] = adj_mat: unused by the reference -> never read (saves 268MB) */
    const float* W_tp  = (const float*)d_in[3];
    const float* W1    = (const float*)d_in[4];
    const float* b1    = (const float*)d_in[5];
    const float* W2    = (const float*)d_in[6];
    const float* b2    = (const float*)d_in[7];

    float* out = (float*)d_out;
    float* ge  = (float*)d_ws;

    dipole_reduce_kernel<<<B_DIM, 256, 0, stream>>>(feats, coors, W_tp, ge);
    dipole_mlp_wmma_kernel<<<(B_DIM / 16) / 4, 128, 0, stream>>>(ge, W1, b1, W2, b2, out);
}